// PepEmbeeding_42700564857378
// MI455X (gfx1250) — compile-verified
//
#include <hip/hip_runtime.h>

// PEP embedding: out[b,h,:] = soft_threshold(emb[x[b,h],:], s[x[b,h],:])
// Gather-then-threshold (NOT threshold-then-gather): ~1.26 GB traffic, ~54us
// roofline at 23.3 TB/s. Pure bandwidth problem -> async-to-LDS gather pipeline
// (global_load_async_to_lds_b128 + s_wait_asynccnt keeps 16 512B gather
// requests in flight per wave with zero VGPR cost for load data).

#define HIDDEN 128

constexpr int kWavesPerBlock = 8;     // 256 threads = 8 wave32
constexpr int kBlockSize     = 256;
constexpr int kDepth         = 4;     // rows per pipeline chunk (8 async b128 in flight/chunk)
constexpr int kGridBlocks    = 2048;  // fixed grid -> nwaves is a compile-time power of two
constexpr int kNWaves        = kGridBlocks * kWavesPerBlock;  // 16384

#if __has_builtin(__builtin_amdgcn_global_load_async_to_lds_b128)
#define PEP_ASYNC 1
#else
#define PEP_ASYNC 0
#endif

// Builtin parameter types (confirmed by clang diagnostic): vector_size(16) int
// pointers, global (as1) source and LDS (as3) destination.
typedef int v4i __attribute__((vector_size(16)));
typedef __attribute__((address_space(1))) v4i* global_v4i_ptr;
typedef __attribute__((address_space(3))) v4i* lds_v4i_ptr;

// Inline-asm wait: "memory" clobber guarantees the compiler cannot hoist the
// LDS reads above the async-counter wait.
#define WAIT_ASYNC(n) asm volatile("s_wait_asynccnt %0" ::"i"(n) : "memory")

__device__ __forceinline__ float soft_thr(float v, float sv) {
  // sigmoid via fast exp + fast reciprocal (v_exp_f32 / v_rcp_f32)
  float sig = __fdividef(1.0f, 1.0f + __expf(-sv));
  float m = fmaxf(fabsf(v) - sig, 0.0f);
  // relu output >= 0, so copysign reproduces sign(v)*m exactly
  // (v==0 -> m==0 since sigmoid>0, so the sign-of-zero case is exact-zero anyway)
  return copysignf(m, v);
}

__global__ __launch_bounds__(kBlockSize) void pep_gather_kernel(
    const int* __restrict__ x, const float* __restrict__ emb,
    const float* __restrict__ s, float* __restrict__ out, int rows) {
  const int lane = threadIdx.x & 31;
  const int wib  = threadIdx.x >> 5;
  const int gwave = blockIdx.x * kWavesPerBlock + wib;

#if PEP_ASYNC
  // Per-wave: 2 chunk buffers x kDepth rows x (128 emb + 128 s) floats = 8KB
  __shared__ float lds[kWavesPerBlock * 2 * kDepth * 2 * HIDDEN];  // 64 KB/WG
  float* wbase = &lds[wib * (2 * kDepth * 2 * HIDDEN)];

  // kNWaves is a compile-time power of two: this is a shift, not a division.
  int cnt = 0;
  if (gwave < rows) cnt = (rows - 1 - gwave) / kNWaves + 1;

  auto issue = [&](int j, int buf) {
    const int r = gwave + j * kNWaves;
    // r is wave-uniform; readfirstlane makes idx an SGPR so the row base
    // address math is scalar and async loads use saddr + lane-offset form.
    const int idx = __builtin_amdgcn_readfirstlane(x[r]);
    const float* ep = emb + (size_t)idx * HIDDEN + lane * 4;
    const float* sp = s   + (size_t)idx * HIDDEN + lane * 4;
    float* slot = wbase + (buf * kDepth + (j & (kDepth - 1))) * (2 * HIDDEN);
    __builtin_amdgcn_global_load_async_to_lds_b128(
        (global_v4i_ptr)(ep), (lds_v4i_ptr)(slot + lane * 4), 0, 0);
    __builtin_amdgcn_global_load_async_to_lds_b128(
        (global_v4i_ptr)(sp), (lds_v4i_ptr)(slot + HIDDEN + lane * 4), 0, 0);
  };

  // Prologue: fill chunk buffer 0.
  const int n0 = cnt < kDepth ? cnt : kDepth;
  for (int k = 0; k < n0; ++k) issue(k, 0);

  int buf = 0;
  for (int j = 0; j < cnt; j += kDepth) {
    int ncur = cnt - j;
    if (ncur > kDepth) ncur = kDepth;
    const int jn = j + kDepth;
    int nnext = cnt - jn;
    if (nnext < 0) nnext = 0;
    if (nnext > kDepth) nnext = kDepth;

    // Issue next chunk into the other buffer before consuming this one.
    for (int k = 0; k < nnext; ++k) issue(jn + k, buf ^ 1);

    // In-order completion: outstanding <= 2*kDepth means the 2*ncur oldest
    // (current chunk) have landed in LDS. Tail chunks drain fully.
    if (nnext == kDepth) {
      WAIT_ASYNC(2 * kDepth);
    } else {
      WAIT_ASYNC(0);
    }

    for (int k = 0; k < ncur; ++k) {
      const int r = gwave + (j + k) * kNWaves;
      float* slot = wbase + (buf * kDepth + k) * (2 * HIDDEN);
      float4 v = *(float4*)(slot + lane * 4);             // ds_load_b128
      float4 t = *(float4*)(slot + HIDDEN + lane * 4);    // ds_load_b128
      float4 o;
      o.x = soft_thr(v.x, t.x);
      o.y = soft_thr(v.y, t.y);
      o.z = soft_thr(v.z, t.z);
      o.w = soft_thr(v.w, t.w);
      *(float4*)(out + (size_t)r * HIDDEN + lane * 4) = o;  // global_store_b128
    }
    buf ^= 1;
  }
#else
  // Fallback: direct b128 gather (loadcnt pipelined by the compiler).
  for (int r = gwave; r < rows; r += kNWaves) {
    const int idx = __builtin_amdgcn_readfirstlane(x[r]);
    float4 v = *(const float4*)(emb + (size_t)idx * HIDDEN + lane * 4);
    float4 t = *(const float4*)(s   + (size_t)idx * HIDDEN + lane * 4);
    float4 o;
    o.x = soft_thr(v.x, t.x);
    o.y = soft_thr(v.y, t.y);
    o.z = soft_thr(v.z, t.z);
    o.w = soft_thr(v.w, t.w);
    *(float4*)(out + (size_t)r * HIDDEN + lane * 4) = o;
  }
#endif
}

extern "C" void kernel_launch(void* const* d_in, const int* in_sizes, int n_in,
                              void* d_out, int out_size, void* d_ws, size_t ws_size,
                              hipStream_t stream) {
  const int*   x   = (const int*)d_in[0];     // [16384, 50] int32
  const float* emb = (const float*)d_in[1];   // [1e6, 128] f32
  const float* s   = (const float*)d_in[2];   // [1e6, 128] f32
  float*       out = (float*)d_out;           // [16384, 50, 128] f32

  const int rows = in_sizes[0];               // 16384 * 50 = 819200
  pep_gather_kernel<<<kGridBlocks, kBlockSize, 0, stream>>>(x, emb, s, out, rows);
}